// DeepWalk_31026843747208
// MI455X (gfx1250) — compile-verified
//
#include <hip/hip_runtime.h>
#include <math.h>

#define EMD_DIM 128

typedef float v2f __attribute__((ext_vector_type(2)));
typedef float v8f __attribute__((ext_vector_type(8)));

// Single-wave latency-optimized hierarchical-softmax path kernel.
//  - one wave32, two 16-row WMMA tiles cover the <=20 path rows
//  - V_WMMA_F32_16X16X4_F32 keeps full f32 precision (product of ~20
//    logsigmoids amplifies relative error, so no f16 inputs)
__global__ __launch_bounds__(32, 1)
void deepwalk_hs_kernel(const int* __restrict__ vjp,
                        const int* __restrict__ ukp,
                        const float* __restrict__ emd,
                        const float* __restrict__ hsw,
                        float* __restrict__ out,
                        int num_V)
{
    const int lane = threadIdx.x;       // 0..31 (wave32)
    const int m    = lane & 15;         // row within a 16-row tile
    const int koff = (lane >> 4) * 2;   // K sub-offset inside each K=4 chunk

    // ---- uniform scalars (round trip #1) ----
    const int v_j = vjp[0];
    const int u_k = ukp[0];

    // ---- heap ancestor path of leaf (num_V-1+u_k), leaf-to-root incl. root 0 ----
    // Uniform trip counts -> no EXEC divergence; per-lane selects pick the two
    // row indices this lane serves (tile0 row m, tile1 row m+16).
    long long n = (long long)(num_V - 1) + (long long)u_k;
    int idx0 = 0, idx1 = 0, L = 0;
#pragma unroll
    for (int i = 0; i < 24; ++i) {
        if (n > 0) {
            n = (n - 1) >> 1;
            if (i == m)      idx0 = (int)n;
            if (i == m + 16) idx1 = (int)n;
            ++L;
        }
    }

    const float* __restrict__ e  = emd + (size_t)v_j  * EMD_DIM;
    const float* __restrict__ h0 = hsw + (size_t)idx0 * EMD_DIM;
    const float* __restrict__ h1 = hsw + (size_t)idx1 * EMD_DIM;

    // ---- dots via V_WMMA_F32_16X16X4_F32, K=128 in 32 chunks of 4 ----
    // A layout (16x4 f32): lanes 0-15 hold K={0,1}, lanes 16-31 K={2,3}.
    // B = embedding chunk replicated across all 16 columns, so D[:,0] = H.e.
    v8f acc0 = {};
    v8f acc1 = {};
#pragma unroll
    for (int c = 0; c < EMD_DIM / 4; ++c) {
        const int k = c * 4 + koff;
        v2f a0 = { h0[k], h0[k + 1] };
        v2f a1 = { h1[k], h1[k + 1] };
        v2f bb = { e[k],  e[k + 1]  };
        acc0 = __builtin_amdgcn_wmma_f32_16x16x4_f32(
                   false, a0, false, bb, (short)0, acc0, false, false);
        acc1 = __builtin_amdgcn_wmma_f32_16x16x4_f32(
                   false, a1, false, bb, (short)0, acc1, false, false);
    }

    // ---- extract D column N=0 -> y[0..31] in LDS ----
    // D layout (32-bit 16x16): lane 0 holds M=0..7 in acc[0..7] (N=0),
    // lane 16 holds M=8..15 (N=0).
    __shared__ float sy[32];
    if (lane == 0) {
#pragma unroll
        for (int r = 0; r < 8; ++r) { sy[r]      = acc0[r]; sy[16 + r] = acc1[r]; }
    }
    if (lane == 16) {
#pragma unroll
        for (int r = 0; r < 8; ++r) { sy[8 + r]  = acc0[r]; sy[24 + r] = acc1[r]; }
    }
    __syncthreads();

    // ---- per-lane log_sigmoid, masked product over the L real path rows ----
    float x  = sy[lane];
    float ls = fminf(x, 0.0f) - log1pf(expf(-fabsf(x)));
    float p  = (lane < L) ? ls : 1.0f;

    // wave32 product reduction
#pragma unroll
    for (int off = 16; off > 0; off >>= 1)
        p *= __shfl_xor(p, off, 32);

    if (lane == 0)
        out[0] = -p;
}

extern "C" void kernel_launch(void* const* d_in, const int* in_sizes, int n_in,
                              void* d_out, int out_size, void* d_ws, size_t ws_size,
                              hipStream_t stream) {
    (void)d_ws; (void)ws_size; (void)out_size; (void)n_in;
    const int*   v_j = (const int*)d_in[0];
    const int*   u_k = (const int*)d_in[1];
    const float* emd = (const float*)d_in[2];
    const float* hsw = (const float*)d_in[3];
    float*       out = (float*)d_out;
    const int num_V  = in_sizes[2] / EMD_DIM;

    deepwalk_hs_kernel<<<1, 32, 0, stream>>>(v_j, u_k, emd, hsw, out, num_V);
}